// BoundingBoxDiscipline_14413910245512
// MI455X (gfx1250) — compile-verified
//
#include <hip/hip_runtime.h>
#include <math.h>

// ---------------- CDNA5 async-path detection ----------------
#if __has_builtin(__builtin_amdgcn_global_load_async_to_lds_b128)
#define USE_ASYNC_LDS 1
#pragma message("PATH: ASYNC double-buffered global_load_async_to_lds_b128")
#else
#define USE_ASYNC_LDS 0
#pragma message("PATH: DIRECT (async builtin missing)")
#endif
#if __has_builtin(__builtin_amdgcn_s_wait_asynccnt)
#define HAVE_WAIT_ASYNC_BUILTIN 1
#else
#define HAVE_WAIT_ASYNC_BUILTIN 0
#endif

// ---- problem constants ----
#define BATCHES 16
#define Hd 512
#define Wd 512
#define Cd 21
#define PIX_PER_BATCH   (Hd * Wd)                      // 262144
#define FLOATS_PER_BATCH ((size_t)PIX_PER_BATCH * Cd)  // 5,505,024
#define BLOCKS_PER_BATCH 64
#define NTHREADS 256
#define PIX_PER_BLOCK (PIX_PER_BATCH / BLOCKS_PER_BATCH)  // 4096
#define BIGI (1 << 30)

// ---- double-buffered async tiling ----
#define TILE_PIX 128
#define TILE_AB_BYTES (TILE_PIX * Cd * 4)          // 10752 B per array per tile
#define TILE_AB_FLOATS (TILE_PIX * Cd)             // 2688 floats
#define TILE_BUF_BYTES (2 * TILE_AB_BYTES)         // 21504 B per buffer (pred+true)
#define TILE_BUF_FLOATS (2 * TILE_AB_FLOATS)       // 5376 floats
#define TILE_XFERS_TOT (TILE_BUF_BYTES / 16)       // 1344 b128 transfers per tile
#define NTILES (PIX_PER_BLOCK / TILE_PIX)          // 32 tiles per block

// workspace layout per batch (8 ints):
// [0]=p_ymin [1]=p_xmin [2]=p_ymax [3]=p_xmax [4]=t_ymin [5]=t_xmin [6]=t_ymax [7]=t_xmax
// slots with (slot & 2)==0 are "min" slots.

__global__ void bbox_init_ws(int* __restrict__ ws) {
    int s = threadIdx.x;            // 0..127
    ws[s] = ((s & 2) == 0) ? BIGI : -BIGI;
}

#if USE_ASYNC_LDS
typedef int v4i __attribute__((ext_vector_type(4)));
typedef __attribute__((address_space(1))) v4i* as1_v4ip;
typedef __attribute__((address_space(3))) v4i* as3_v4ip;

__device__ __forceinline__ void async_copy_b128(const void* g, void* l) {
    // each lane copies 16 bytes from global address g to LDS address l, tracked by ASYNCcnt
    __builtin_amdgcn_global_load_async_to_lds_b128((as1_v4ip)g, (as3_v4ip)l, 0, 0);
}
__device__ __forceinline__ void wait_async_le(int) {}
#if HAVE_WAIT_ASYNC_BUILTIN
#define WAIT_ASYNC(N) __builtin_amdgcn_s_wait_asynccnt(N)
#else
#define WAIT_ASYNC(N) asm volatile("s_wait_asynccnt " #N ::: "memory")
#endif

// Issue all 1344 async b128 transfers for one tile.
// Waves 0-1 (tid<64) issue 6 instructions, waves 2-7 issue 5 (wave-uniform).
__device__ __forceinline__ void issue_tile(const char* psrc, const char* tsrc,
                                           char* buf, int tid) {
#pragma unroll
    for (int i = 0; i < 6; ++i) {
        const int k = tid + i * NTHREADS;
        if (k < TILE_XFERS_TOT) {
            const int b = k * 16;
            const char* g = (b < TILE_AB_BYTES) ? (psrc + b)
                                                : (tsrc + (b - TILE_AB_BYTES));
            async_copy_b128(g, buf + b);
        }
    }
}
#endif

__device__ __forceinline__ float max21_global(const float* __restrict__ p) {
    float m = p[20];
#pragma unroll
    for (int k = 0; k < 5; ++k) {
        float4 v = *(const float4*)(p + 4 * k);
        m = fmaxf(m, fmaxf(fmaxf(v.x, v.y), fmaxf(v.z, v.w)));
    }
    return m;
}

__global__ __launch_bounds__(NTHREADS) void bbox_reduce(
        const float* __restrict__ pred, const float* __restrict__ tru,
        int* __restrict__ ws) {
    const int batch = blockIdx.x / BLOCKS_PER_BATCH;
    const int chunk = blockIdx.x % BLOCKS_PER_BATCH;
    const int tid = threadIdx.x;

    int pymin = BIGI, pxmin = BIGI, pymax = -BIGI, pxmax = -BIGI;
    int tymin = BIGI, txmin = BIGI, tymax = -BIGI, txmax = -BIGI;

#if USE_ASYNC_LDS
    // two buffers; each: pred floats [0, TILE_AB_FLOATS), true [TILE_AB_FLOATS, ...)
    __shared__ float stage[2 * TILE_BUF_FLOATS];   // 43008 bytes

    const char* predB = (const char*)(pred + (size_t)batch * FLOATS_PER_BATCH);
    const char* truB  = (const char*)(tru  + (size_t)batch * FLOATS_PER_BATCH);
    char* const buf0 = (char*)stage;
    char* const buf1 = (char*)stage + TILE_BUF_BYTES;

    const size_t blockByte = (size_t)(chunk * PIX_PER_BLOCK) * (Cd * 4);

    // prologue: stream tile 0 into buffer 0
    issue_tile(predB + blockByte, truB + blockByte, buf0, tid);

    for (int t = 0; t < NTILES; ++t) {
        if (t + 1 < NTILES) {
            // overlap: stream tile t+1 into the other buffer
            const size_t nb = blockByte + (size_t)(t + 1) * TILE_AB_BYTES;
            issue_tile(predB + nb, truB + nb, ((t + 1) & 1) ? buf1 : buf0, tid);
            // exact-count wait: async loads complete in order, so ASYNCcnt <=
            // (#ops just issued by this wave) => tile t fully resident
            if (tid < 64) { WAIT_ASYNC(6); } else { WAIT_ASYNC(5); }
        } else {
            WAIT_ASYNC(0);
        }
        __syncthreads();     // all waves' tile-t transfers complete

        {
            const float* bufF = (const float*)((t & 1) ? buf1 : buf0);
            const int p    = tid >> 1;        // pixel within tile (lane pairs share)
            const int half = tid & 1;         // 0: channels 0..10, 1: channels 11..20
            const float* lp = bufF + p * Cd;
            const float* lt = bufF + TILE_AB_FLOATS + p * Cd;

            const int c0 = half ? 11 : 0;
            float pm = lp[c0], tm = lt[c0];
#pragma unroll
            for (int j = 1; j < 10; ++j) {
                pm = fmaxf(pm, lp[c0 + j]);
                tm = fmaxf(tm, lt[c0 + j]);
            }
            if (!half) { pm = fmaxf(pm, lp[10]); tm = fmaxf(tm, lt[10]); }
            // combine halves; both lanes get the full-record max
            pm = fmaxf(pm, __shfl_xor(pm, 1, 32));
            tm = fmaxf(tm, __shfl_xor(tm, 1, 32));

            const int pixel = chunk * PIX_PER_BLOCK + t * TILE_PIX + p;
            const int y = pixel >> 9;
            const int x = pixel & (Wd - 1);
            // duplicate updates from lane pairs are harmless (min/max idempotent)
            if (pm > 0.3f) {
                pymin = min(pymin, y); pxmin = min(pxmin, x);
                pymax = max(pymax, y); pxmax = max(pxmax, x);
            }
            if (tm > 0.5f) {
                tymin = min(tymin, y); txmin = min(txmin, x);
                tymax = max(tymax, y); txmax = max(txmax, x);
            }
        }
        __syncthreads();     // reads done before this buffer is refilled (t+2)
    }
#else
    const size_t batchBase = (size_t)batch * FLOATS_PER_BATCH;
    const int pix0 = chunk * PIX_PER_BLOCK + tid;

#pragma unroll 4
    for (int i = 0; i < PIX_PER_BLOCK / NTHREADS; ++i) {
        const int pixel = pix0 + i * NTHREADS;
        const float* pp = pred + batchBase + (size_t)pixel * Cd;
        const float* tp = tru  + batchBase + (size_t)pixel * Cd;
        __builtin_prefetch(pp + NTHREADS * Cd, 0, 1);
        __builtin_prefetch(tp + NTHREADS * Cd, 0, 1);
        const float pm = max21_global(pp);
        const float tm = max21_global(tp);
        const int y = pixel >> 9;
        const int x = pixel & (Wd - 1);
        if (pm > 0.3f) {
            pymin = min(pymin, y); pxmin = min(pxmin, x);
            pymax = max(pymax, y); pxmax = max(pxmax, x);
        }
        if (tm > 0.5f) {
            tymin = min(tymin, y); txmin = min(txmin, x);
            tymax = max(tymax, y); txmax = max(txmax, x);
        }
    }
#endif

    // block reduction via LDS atomics (ds_min_i32 / ds_max_i32)
    __shared__ int sh[8];
    if (tid < 8) sh[tid] = ((tid & 2) == 0) ? BIGI : -BIGI;
    __syncthreads();

    atomicMin(&sh[0], pymin); atomicMin(&sh[1], pxmin);
    atomicMax(&sh[2], pymax); atomicMax(&sh[3], pxmax);
    atomicMin(&sh[4], tymin); atomicMin(&sh[5], txmin);
    atomicMax(&sh[6], tymax); atomicMax(&sh[7], txmax);
    __syncthreads();

    if (tid < 8) {
        int* dst = ws + batch * 8 + tid;
        if ((tid & 2) == 0) atomicMin(dst, sh[tid]);
        else                atomicMax(dst, sh[tid]);
    }
}

__global__ void bbox_finalize(const int* __restrict__ ws, float* __restrict__ out) {
    const int lane = threadIdx.x;   // one wave32; lanes 0..15 carry batches
    float pen = 0.0f;
    if (lane < BATCHES) {
        const int* w = ws + lane * 8;
        const int py1 = w[0], px1 = w[1], py2 = w[2], px2 = w[3];
        const int ty1 = w[4], tx1 = w[5], ty2 = w[6], tx2 = w[7];
        const bool valid = (py1 != BIGI) && (ty1 != BIGI);
        if (valid) {
            const float pa = (float)((py2 - py1 + 1) * (px2 - px1 + 1));
            const float ta = (float)((ty2 - ty1 + 1) * (tx2 - tx1 + 1));
            const float area_pen = fmaxf(pa - ta, 0.0f) / (ta + 1.0f);
            const float pcy = (float)(py1 + py2) * 0.5f;
            const float pcx = (float)(px1 + px2) * 0.5f;
            const float tcy = (float)(ty1 + ty2) * 0.5f;
            const float tcx = (float)(tx1 + tx2) * 0.5f;
            const float dy = pcy - tcy, dx = pcx - tcx;
            pen = area_pen + sqrtf(dy * dy + dx * dx) * 0.05f;  // /20
        } else {
            pen = 1.0f;
        }
    }
    for (int o = 8; o > 0; o >>= 1) pen += __shfl_down(pen, o, 32);
    if (lane == 0) out[0] = 0.05f * pen * (1.0f / 16.0f);   // PENALTY_WEIGHT * mean
}

extern "C" void kernel_launch(void* const* d_in, const int* in_sizes, int n_in,
                              void* d_out, int out_size, void* d_ws, size_t ws_size,
                              hipStream_t stream) {
    const float* pred = (const float*)d_in[0];   // prediction_probs, fp32
    const float* tru  = (const float*)d_in[1];   // expected_onehot,  fp32
    float* out = (float*)d_out;
    int* ws = (int*)d_ws;                        // 128 ints used

    bbox_init_ws<<<1, 128, 0, stream>>>(ws);
    bbox_reduce<<<BATCHES * BLOCKS_PER_BATCH, NTHREADS, 0, stream>>>(pred, tru, ws);
    bbox_finalize<<<1, 32, 0, stream>>>(ws, out);
}